// GraphSAGE_197568496007
// MI455X (gfx1250) — compile-verified
//
#include <hip/hip_runtime.h>

typedef __attribute__((ext_vector_type(2))) float v2f;
typedef __attribute__((ext_vector_type(8))) float v8f;
typedef int v4i __attribute__((vector_size(16)));   // matches builtin's b128 pointee

// Problem constants (from the reference)
#define F0   512
#define H1   512
#define H2   256
#define BB   1024
#define NN1  25
#define NN2  10
#define M2   (BB * NN2)      // 10240
#define MT   (BB + M2)       // 11264 rows of the stacked layer-0 GEMM

// GEMM tiling
#define BM   128
#define BN   64
#define BK   32
#define STR  (BK + 4)        // 36-float row stride: 16B-aligned rows, conflict-free frags
#define NTHREADS 256

// ---------------------------------------------------------------------------
// gfx1250 async global->LDS path (ASYNCcnt), probed via __has_builtin with a
// register-staged double-buffer fallback.
// Builtin signature (from hipcc diagnostics):
//   b32 : (as(1) int*  src, as(3) int*  dst, imm int offset, imm int cpol)
//   b128: (as(1) v4i*  src, as(3) v4i*  dst, imm int offset, imm int cpol)
// ---------------------------------------------------------------------------
#if __has_builtin(__builtin_amdgcn_global_load_async_to_lds_b128) && \
    __has_builtin(__builtin_amdgcn_global_load_async_to_lds_b32)
#define SAGE_ASYNC 1
#else
#define SAGE_ASYNC 0
#endif

#if SAGE_ASYNC
__device__ __forceinline__ void async_copy_b128(const float* g, float* l) {
    __builtin_amdgcn_global_load_async_to_lds_b128(
        (__attribute__((address_space(1))) v4i*)(float*)g,
        (__attribute__((address_space(3))) v4i*)l, 0, 0);
}
__device__ __forceinline__ void async_copy_b32(const float* g, float* l) {
    __builtin_amdgcn_global_load_async_to_lds_b32(
        (__attribute__((address_space(1))) int*)(float*)g,
        (__attribute__((address_space(3))) int*)l, 0, 0);
}
__device__ __forceinline__ void wait_async() {
#if __has_builtin(__builtin_amdgcn_s_wait_asynccnt)
    __builtin_amdgcn_s_wait_asynccnt(0);
#else
    asm volatile("s_wait_asynccnt 0x0" ::: "memory");
#endif
}
#endif

// -----------------------------------------------------------------------------
// Kernel 1: gather + mean-aggregate + concat  ->  Xcat [MT, 2*F0]
//   rows 0..1023 : [ x[b]             | mean_j feats[neigh1[b,j]]   ]
//   rows 1024..  : [ feats[neigh2[m]] | mean_j feats[neigh2_1[m,j]] ]
// -----------------------------------------------------------------------------
__global__ __launch_bounds__(256) void sage_gather_concat(
    const float* __restrict__ feats,
    const float* __restrict__ x,
    const int*   __restrict__ neigh1,
    const int*   __restrict__ neigh2,     // flat [B*N2]
    const int*   __restrict__ neigh2_1,   // [B*N2, N1]
    float*       __restrict__ Xcat)       // [MT, 2*F0]
{
    const int row = blockIdx.x;
    const int c0  = threadIdx.x * 2;      // 2 consecutive floats -> coalesced

    const float* selfp;
    const int*   nidx;
    if (row < BB) {
        selfp = x + (size_t)row * F0 + c0;
        nidx  = neigh1 + (size_t)row * NN1;
    } else {
        const int m = row - BB;
        selfp = feats + (size_t)neigh2[m] * F0 + c0;
        nidx  = neigh2_1 + (size_t)m * NN1;
    }

    const float s0 = selfp[0];
    const float s1 = selfp[1];

    float a0 = 0.f, a1 = 0.f;
    const float* fr = feats + (size_t)nidx[0] * F0 + c0;
    #pragma unroll 5
    for (int j = 0; j < NN1; ++j) {
        const float* cur = fr;
        if (j + 1 < NN1) {
            fr = feats + (size_t)nidx[j + 1] * F0 + c0;
            __builtin_prefetch(fr, 0, 1);   // global_prefetch_b8: hide gather latency
        }
        a0 += cur[0];
        a1 += cur[1];
    }
    const float inv = 1.0f / (float)NN1;

    float* outp = Xcat + (size_t)row * (2 * F0);
    outp[c0]          = s0;
    outp[c0 + 1]      = s1;
    outp[F0 + c0]     = a0 * inv;
    outp[F0 + c0 + 1] = a1 * inv;
}

// -----------------------------------------------------------------------------
// Kernel 3: layer-1 mean over the 10 neighbor embeddings + concat -> X3
// -----------------------------------------------------------------------------
__global__ __launch_bounds__(256) void sage_layer1_concat(
    const float* __restrict__ Hcat,   // [MT, H1]
    float*       __restrict__ X3)     // [BB, 2*H1]
{
    const int b  = blockIdx.x;
    const int c0 = threadIdx.x * 2;

    const float* sp = Hcat + (size_t)b * H1 + c0;
    const float s0 = sp[0];
    const float s1 = sp[1];

    float a0 = 0.f, a1 = 0.f;
    #pragma unroll
    for (int n = 0; n < NN2; ++n) {
        const float* hp = Hcat + (size_t)(BB + b * NN2 + n) * H1 + c0;
        a0 += hp[0];
        a1 += hp[1];
    }
    const float inv = 1.0f / (float)NN2;

    float* outp = X3 + (size_t)b * (2 * H1);
    outp[c0]          = s0;
    outp[c0 + 1]      = s1;
    outp[H1 + c0]     = a0 * inv;
    outp[H1 + c0 + 1] = a1 * inv;
}

// -----------------------------------------------------------------------------
// WMMA f32 GEMM:  C = relu(A @ B + bias)
//   A [M,K] row-major, B [K,N] row-major, bias [N], C [M,N]
// Block = 256 threads = 8 waves. Tile BM=128 x BN=64, BK=32, double-buffered.
// Wave owns a 32x32 region = 2x2 subtiles of 16x16: 4 WMMAs per 4 ds_load_b64.
// Bs is stored TRANSPOSED ([col][k]) so B fragments are contiguous b64 loads.
// Requires M%128==0, N%64==0, K%32==0 (true for all call sites here).
// -----------------------------------------------------------------------------
__global__ __launch_bounds__(NTHREADS) void wmma_gemm_bias_relu(
    const float* __restrict__ A,
    const float* __restrict__ B,
    const float* __restrict__ bias,
    float*       __restrict__ C,
    int M, int N, int K)
{
    __shared__ float As[2][BM * STR];   // [row][k]
    __shared__ float Bs[2][BN * STR];   // [col][k]  (transposed)

    const int tid  = threadIdx.x;
    const int lane = tid & 31;
    const int wave = tid >> 5;
    const int half = lane >> 4;         // lanes 16-31 carry K+2/K+3 (ISA layout)
    const int l15  = lane & 15;

    const int blockM = blockIdx.y * BM;
    const int blockN = blockIdx.x * BN;

    const int mBase = (wave & 3) * 32;  // wave's 32-row region inside the tile
    const int nBase = (wave >> 2) * 32; // wave's 32-col region inside the tile
    const int kOff  = half * 2;
    const int aRow0 = mBase + l15, aRow1 = aRow0 + 16;
    const int bCol0 = nBase + l15, bCol1 = bCol0 + 16;

    v8f acc00 = {}, acc01 = {}, acc10 = {}, acc11 = {};

    const int nChunks = K / BK;
    int buf = 0;

#if SAGE_ASYNC
    // ---- async global->LDS tile fill (ASYNCcnt) ----
    auto issue_chunk = [&](int k0, int b) {
        float* ab = &As[b][0];
        float* bb = &Bs[b][0];
        #pragma unroll
        for (int i = 0; i < 4; ++i) {           // A: 128x32 floats as b128/lane
            const int idx = tid + i * NTHREADS; // 0..1023
            const int r = idx >> 3, c = (idx & 7) * 4;
            async_copy_b128(A + (size_t)(blockM + r) * K + (k0 + c),
                            ab + r * STR + c);
        }
        #pragma unroll
        for (int i = 0; i < 8; ++i) {           // B: 32x64 floats, transposed, b32/lane
            const int idx = tid + i * NTHREADS; // 0..2047
            const int r = idx >> 6, c = idx & 63;
            async_copy_b32(B + (size_t)(k0 + r) * N + (blockN + c),
                           bb + c * STR + r);
        }
    };
#else
    // ---- register-staged double buffering fallback ----
    float4 aReg[4];
    float4 bReg[2];
    auto load_regs = [&](int k0) {
        #pragma unroll
        for (int i = 0; i < 4; ++i) {
            const int idx = tid + i * NTHREADS;
            const int r = idx >> 3, c = (idx & 7) * 4;
            aReg[i] = *(const float4*)(A + (size_t)(blockM + r) * K + (k0 + c));
        }
        #pragma unroll
        for (int i = 0; i < 2; ++i) {
            const int idx = tid + i * NTHREADS;   // 0..511 -> 32 rows x 16 float4
            const int r = idx >> 4, c = (idx & 15) * 4;
            bReg[i] = *(const float4*)(B + (size_t)(k0 + r) * N + (blockN + c));
        }
    };
    auto store_regs = [&](int b) {
        #pragma unroll
        for (int i = 0; i < 4; ++i) {
            const int idx = tid + i * NTHREADS;
            const int r = idx >> 3, c = (idx & 7) * 4;
            *(float4*)&As[b][r * STR + c] = aReg[i];
        }
        #pragma unroll
        for (int i = 0; i < 2; ++i) {             // transpose into Bs
            const int idx = tid + i * NTHREADS;
            const int r = idx >> 4, c = (idx & 15) * 4;
            Bs[b][(c + 0) * STR + r] = bReg[i].x;
            Bs[b][(c + 1) * STR + r] = bReg[i].y;
            Bs[b][(c + 2) * STR + r] = bReg[i].z;
            Bs[b][(c + 3) * STR + r] = bReg[i].w;
        }
    };
#endif

    auto compute = [&](int b) {
        #pragma unroll
        for (int kk = 0; kk < BK; kk += 4) {
            const int ko = kk + kOff;
            const v2f a0 = *(const v2f*)&As[b][aRow0 * STR + ko];
            const v2f a1 = *(const v2f*)&As[b][aRow1 * STR + ko];
            const v2f b0 = *(const v2f*)&Bs[b][bCol0 * STR + ko];
            const v2f b1 = *(const v2f*)&Bs[b][bCol1 * STR + ko];
            acc00 = __builtin_amdgcn_wmma_f32_16x16x4_f32(false, a0, false, b0,
                                                          (short)0, acc00, false, false);
            acc01 = __builtin_amdgcn_wmma_f32_16x16x4_f32(false, a0, false, b1,
                                                          (short)0, acc01, false, false);
            acc10 = __builtin_amdgcn_wmma_f32_16x16x4_f32(false, a1, false, b0,
                                                          (short)0, acc10, false, false);
            acc11 = __builtin_amdgcn_wmma_f32_16x16x4_f32(false, a1, false, b1,
                                                          (short)0, acc11, false, false);
        }
    };

#if SAGE_ASYNC
    issue_chunk(0, 0);
    wait_async();
    __syncthreads();
    for (int ci = 0; ci < nChunks; ++ci) {
        if (ci + 1 < nChunks) issue_chunk((ci + 1) * BK, buf ^ 1);
        compute(buf);
        wait_async();
        __syncthreads();
        buf ^= 1;
    }
#else
    load_regs(0);
    store_regs(0);
    __syncthreads();
    for (int ci = 0; ci < nChunks; ++ci) {
        const bool more = (ci + 1 < nChunks);
        if (more) load_regs((ci + 1) * BK);   // global loads fly over the WMMAs
        compute(buf);
        if (more) store_regs(buf ^ 1);
        __syncthreads();
        buf ^= 1;
    }
#endif

    // ---- epilogue: bias + relu; C/D layout: VGPR i, lanes 0-15 -> row i,
    //      lanes 16-31 -> row i+8, N = lane&15 ----
    const int r0  = blockM + mBase + half * 8;
    const int r1  = r0 + 16;
    const int gc0 = blockN + nBase + l15;
    const int gc1 = gc0 + 16;
    const float bi0 = bias[gc0];
    const float bi1 = bias[gc1];
    #pragma unroll
    for (int i = 0; i < 8; ++i) {
        C[(size_t)(r0 + i) * N + gc0] = fmaxf(acc00[i] + bi0, 0.f);
        C[(size_t)(r0 + i) * N + gc1] = fmaxf(acc01[i] + bi1, 0.f);
        C[(size_t)(r1 + i) * N + gc0] = fmaxf(acc10[i] + bi0, 0.f);
        C[(size_t)(r1 + i) * N + gc1] = fmaxf(acc11[i] + bi1, 0.f);
    }
}

// -----------------------------------------------------------------------------
// Host launcher
// Inputs (setup_inputs order):
//  0 feats  1 x  2 W0  3 b0  4 W1  5 b1  6 nodes  7 neigh1  8 neigh2  9 neigh2_1
// Output: [1024,256] f32
// -----------------------------------------------------------------------------
extern "C" void kernel_launch(void* const* d_in, const int* in_sizes, int n_in,
                              void* d_out, int out_size, void* d_ws, size_t ws_size,
                              hipStream_t stream) {
    (void)in_sizes; (void)n_in; (void)out_size;

    const float* feats    = (const float*)d_in[0];
    const float* x        = (const float*)d_in[1];
    const float* W0       = (const float*)d_in[2];
    const float* b0       = (const float*)d_in[3];
    const float* W1       = (const float*)d_in[4];
    const float* b1       = (const float*)d_in[5];
    const int*   neigh1   = (const int*)d_in[7];
    const int*   neigh2   = (const int*)d_in[8];
    const int*   neigh2_1 = (const int*)d_in[9];
    float*       out      = (float*)d_out;

    const size_t szXcat = (size_t)MT * (2 * F0) * sizeof(float);   // 46.1 MB
    const size_t szHcat = (size_t)MT * H1 * sizeof(float);         // 23.1 MB
    const size_t szX3   = (size_t)BB * (2 * H1) * sizeof(float);   //  4.2 MB
    auto align256 = [](size_t v) { return (v + 255) & ~(size_t)255; };
    if (ws_size < align256(szXcat) + align256(szHcat) + align256(szX3)) return;

    char* wsp   = (char*)d_ws;
    float* Xcat = (float*)wsp;                       wsp += align256(szXcat);
    float* Hcat = (float*)wsp;                       wsp += align256(szHcat);
    float* X3   = (float*)wsp;

    // 1) gather + mean + concat -> Xcat [11264, 1024]
    sage_gather_concat<<<MT, 256, 0, stream>>>(feats, x, neigh1, neigh2, neigh2_1, Xcat);

    // 2) Hcat = relu(Xcat @ W0 + b0)  [11264, 512]
    wmma_gemm_bias_relu<<<dim3(H1 / BN, MT / BM), NTHREADS, 0, stream>>>(
        Xcat, W0, b0, Hcat, MT, H1, 2 * F0);

    // 3) layer-1 mean + concat -> X3 [1024, 1024]
    sage_layer1_concat<<<BB, 256, 0, stream>>>(Hcat, X3);

    // 4) out = relu(X3 @ W1 + b1)  [1024, 256]
    wmma_gemm_bias_relu<<<dim3(H2 / BN, BB / BM), NTHREADS, 0, stream>>>(
        X3, W1, b1, out, BB, H2, 2 * H1);
}